// UString_62045097558247
// MI455X (gfx1250) — compile-verified
//
#include <hip/hip_runtime.h>

#define B_   16
#define N_   4096
#define E_   131072          // 2^17
#define CIN  256
#define COUT 256

typedef __attribute__((ext_vector_type(2))) float v2f;
typedef __attribute__((ext_vector_type(8))) float v8f;

// ---------------- zero out + deg ----------------
__global__ __launch_bounds__(256) void zero_kernel(float* __restrict__ out,
                                                   float* __restrict__ deg) {
  size_t idx = (size_t)blockIdx.x * 256 + threadIdx.x;
  out[idx] = 0.0f;                       // grid sized exactly to B*N*COUT
  if (idx < (size_t)B_ * N_) deg[idx] = 0.0f;
}

// ---------------- degree scatter ----------------
__global__ __launch_bounds__(256) void degree_kernel(const int* __restrict__ ei,
                                                     const float* __restrict__ ew,
                                                     float* __restrict__ deg) {
  int ge  = blockIdx.x * 256 + threadIdx.x;     // [0, B*E)
  int b   = ge >> 17;                           // E = 2^17
  int e   = ge & (E_ - 1);
  int row = ei[(size_t)b * 2 * E_ + e];
  float w = ew[(size_t)b * E_ + e];
  unsafeAtomicAdd(&deg[b * N_ + row], w);       // global_atomic_add_f32, no return
}

// ---------------- deg -> deg^-1/2 (self loop weight 1 folded in) ----------------
__global__ __launch_bounds__(256) void dinv_kernel(float* __restrict__ deg) {
  int idx = blockIdx.x * 256 + threadIdx.x;
  deg[idx] = rsqrtf(deg[idx] + 1.0f);           // deg+1 > 0 always
}

// ---------------- xw = x @ W via V_WMMA_F32_16X16X4_F32 ----------------
// Block: 256 threads (8 waves). Tile: 128 rows x 64 cols. K chunked by 64.
// LDS layout for W chunk: element (k,n) at ((k>>2)*64 + n)*4 + (k&3)
//  -> a lane's B fragment {K=2h, K=2h+1} at fixed n is a contiguous, bank-
//     conflict-free ds_load_b64.
__global__ __launch_bounds__(256) void gemm_wmma(const float* __restrict__ X,
                                                 const float* __restrict__ W,
                                                 float* __restrict__ XW) {
  __shared__ float Wt[64 * 64];                 // 16 KB
  const int tid    = threadIdx.x;
  const int wave   = tid >> 5;
  const int lane   = tid & 31;
  const int lhalf  = lane >> 4;                 // 0 | 1
  const int l16    = lane & 15;
  const int blockM = blockIdx.x * 128;          // over B*N = 65536 rows
  const int blockN = blockIdx.y * 64;
  const int mBase  = blockM + wave * 16;

  v8f acc0 = {}, acc1 = {}, acc2 = {}, acc3 = {};

  // A fragment source: row (mBase+l16), K components 2*lhalf, 2*lhalf+1
  const float* xrow = X + (size_t)(mBase + l16) * CIN + (lhalf << 1);

  for (int k0 = 0; k0 < CIN; k0 += 64) {
    // stage W[k0..k0+64)[blockN..blockN+64) into LDS (interleaved layout)
#pragma unroll
    for (int i = 0; i < 16; ++i) {
      int idx = tid + 256 * i;                  // 0..4095, coalesced in n
      int k   = idx >> 6;
      int n   = idx & 63;
      Wt[(((k >> 2) * 64 + n) << 2) + (k & 3)] =
          W[(size_t)(k0 + k) * COUT + blockN + n];
    }
    __syncthreads();

#pragma unroll
    for (int kk = 0; kk < 64; kk += 4) {
      v2f a = *(const v2f*)(xrow + k0 + kk);    // global_load_b64
      const float* wb = Wt + (((kk >> 2) * 64) << 2) + (lhalf << 1);
      v2f b0 = *(const v2f*)(wb + ((l16 +  0) << 2));
      v2f b1 = *(const v2f*)(wb + ((l16 + 16) << 2));
      v2f b2 = *(const v2f*)(wb + ((l16 + 32) << 2));
      v2f b3 = *(const v2f*)(wb + ((l16 + 48) << 2));
      acc0 = __builtin_amdgcn_wmma_f32_16x16x4_f32(false, a, false, b0, (short)0, acc0, false, false);
      acc1 = __builtin_amdgcn_wmma_f32_16x16x4_f32(false, a, false, b1, (short)0, acc1, false, false);
      acc2 = __builtin_amdgcn_wmma_f32_16x16x4_f32(false, a, false, b2, (short)0, acc2, false, false);
      acc3 = __builtin_amdgcn_wmma_f32_16x16x4_f32(false, a, false, b3, (short)0, acc3, false, false);
    }
    __syncthreads();
  }

  // C/D layout: VGPR r -> M = r + 8*lhalf, lane l16 -> N = l16 (+16j)
#pragma unroll
  for (int r = 0; r < 8; ++r) {
    size_t o = (size_t)(mBase + 8 * lhalf + r) * COUT + blockN + l16;
    XW[o +  0] = acc0[r];
    XW[o + 16] = acc1[r];
    XW[o + 32] = acc2[r];
    XW[o + 48] = acc3[r];
  }
}

// ---------------- edge scatter: out[row] += norm * xw[col] ----------------
// 32 edges per block, 256 threads = 256 channels -> fully coalesced gather
// and atomic add (1 KB contiguous per edge).
__global__ __launch_bounds__(256) void scatter_kernel(const int* __restrict__ ei,
                                                      const float* __restrict__ ew,
                                                      const float* __restrict__ dinv,
                                                      const float* __restrict__ XW,
                                                      float* __restrict__ out) {
  __shared__ int   s_row[32];
  __shared__ int   s_col[32];
  __shared__ float s_nrm[32];
  const int tid = threadIdx.x;
  const int b   = blockIdx.x >> 12;             // E/32 = 4096 blocks per batch
  if (tid < 32) {
    int e   = ((blockIdx.x & 4095) << 5) + tid;
    int row = ei[(size_t)b * 2 * E_ + e];
    int col = ei[(size_t)b * 2 * E_ + E_ + e];
    float w = ew[(size_t)b * E_ + e];
    s_row[tid] = row;
    s_col[tid] = col;
    s_nrm[tid] = dinv[b * N_ + row] * w * dinv[b * N_ + col];
  }
  __syncthreads();
  const size_t nodeBase = (size_t)b * N_;
#pragma unroll 4
  for (int k = 0; k < 32; ++k) {
    float v = s_nrm[k] * XW[(nodeBase + s_col[k]) * COUT + tid];
    unsafeAtomicAdd(&out[(nodeBase + s_row[k]) * COUT + tid], v);
  }
}

// ---------------- self-loop + ReLU epilogue ----------------
__global__ __launch_bounds__(256) void finalize_kernel(const float* __restrict__ dinv,
                                                       const float* __restrict__ XW,
                                                       float* __restrict__ out) {
  size_t idx  = (size_t)blockIdx.x * 256 + threadIdx.x;
  size_t node = idx >> 8;                       // / COUT
  float  di   = dinv[node];
  float  v    = out[idx] + di * di * XW[idx];
  out[idx]    = fmaxf(v, 0.0f);
}

extern "C" void kernel_launch(void* const* d_in, const int* in_sizes, int n_in,
                              void* d_out, int out_size, void* d_ws, size_t ws_size,
                              hipStream_t stream) {
  const float* x  = (const float*)d_in[0];   // [B, N, CIN]
  const int*   ei = (const int*)  d_in[1];   // [B, 2, E]
  const float* ew = (const float*)d_in[2];   // [B, E]
  const float* W  = (const float*)d_in[3];   // [CIN, COUT]
  float* out = (float*)d_out;                // [B, N, COUT]

  float* xw  = (float*)d_ws;                               // B*N*COUT floats (64 MB)
  float* deg = xw + (size_t)B_ * N_ * COUT;                // B*N floats (deg -> dinv)

  zero_kernel    <<<(B_ * N_ * COUT) / 256, 256, 0, stream>>>(out, deg);
  degree_kernel  <<<(B_ * E_) / 256, 256, 0, stream>>>(ei, ew, deg);
  dinv_kernel    <<<(B_ * N_) / 256, 256, 0, stream>>>(deg);
  gemm_wmma      <<<dim3((B_ * N_) / 128, COUT / 64), 256, 0, stream>>>(x, W, xw);
  scatter_kernel <<<(B_ * E_) / 32, 256, 0, stream>>>(ei, ew, deg, xw, out);
  finalize_kernel<<<(B_ * N_ * COUT) / 256, 256, 0, stream>>>(deg, xw, out);
}